// AEDecoder_45011257262637
// MI455X (gfx1250) — compile-verified
//
#include <hip/hip_runtime.h>
#include <hip/hip_bf16.h>
#include <stdint.h>

typedef float v2f __attribute__((ext_vector_type(2)));
typedef float v4f __attribute__((ext_vector_type(4)));
typedef float v8f __attribute__((ext_vector_type(8)));

#define NUM_TFS   1600
#define NUM_GENES 20000
#define BATCH     128
#define BT        16          // batch rows staged per block
#define FSTRIDE   1604        // padded LDS row stride (floats): conflict-free gather
#define GPB       64          // genes per block (8 waves * 8 iters)
#define PSTRIDE   68          // padded per-gene partial stride (floats)
#define NTHREADS  256

__global__ __launch_bounds__(NTHREADS)
void gene_sparse2_kernel(const float* __restrict__ feat,
                         const float* __restrict__ w1,
                         const float* __restrict__ b1,
                         const float* __restrict__ w2,
                         const float* __restrict__ b2,
                         const int*   __restrict__ cols1,
                         float*       __restrict__ out)
{
    __shared__ float ldsfeat[BT * FSTRIDE];    // [row][tf], padded
    __shared__ float partial[GPB * PSTRIDE];   // [gene_local][n][m], padded

    const int tid    = threadIdx.x;
    const int btile  = blockIdx.y;             // 0..7 (16 rows each)
    const int gblock = blockIdx.x * GPB;

    // ---- Stage features[btile*16 .. +16][0..1600) into LDS (async copy) ----
    {
        const float* gbase = feat + (size_t)(btile * BT) * NUM_TFS;
        for (int row = 0; row < BT; ++row) {
            for (int c4 = tid; c4 < NUM_TFS / 4; c4 += NTHREADS) {
                unsigned lds_off = (unsigned)(uintptr_t)&ldsfeat[row * FSTRIDE + c4 * 4];
                const float* gp  = gbase + (size_t)row * NUM_TFS + c4 * 4;
                asm volatile("global_load_async_to_lds_b128 %0, %1, off"
                             :: "v"(lds_off), "v"(gp) : "memory");
            }
        }
        asm volatile("s_wait_asynccnt 0x0" ::: "memory");
    }
    __syncthreads();

    const int  lane = tid & 31;
    const int  wv   = __builtin_amdgcn_readfirstlane(tid >> 5); // wave id 0..7
    const int  n    = lane & 15;           // M (A rows) / N (B cols) lane index
    const bool half = (lane >> 4) != 0;    // lanes 16..31
    const bool nv   = (n < 4);             // valid B column
    const int  nn   = n & 3;               // clamped column (safe addressing)

    const float* frow = ldsfeat + n * FSTRIDE;

    // ---- Per-gene GEMM via v_wmma_f32_16x16x4_f32, fused bias/leaky/layer2 ----
    for (int i = 0; i < GPB / 8; ++i) {
        const int g  = gblock + i * 8 + wv;         // uniform within wave
        const int gl = i * 8 + wv;                  // gene slot in block
        if (g < NUM_GENES) {
            // 10 uniform TF column indices for this gene (SGPRs)
            const int* cg = cols1 + (size_t)g * 40;
            const int c0 = __builtin_amdgcn_readfirstlane(cg[0]);
            const int c1 = __builtin_amdgcn_readfirstlane(cg[1]);
            const int c2 = __builtin_amdgcn_readfirstlane(cg[2]);
            const int c3 = __builtin_amdgcn_readfirstlane(cg[3]);
            const int c4 = __builtin_amdgcn_readfirstlane(cg[4]);
            const int c5 = __builtin_amdgcn_readfirstlane(cg[5]);
            const int c6 = __builtin_amdgcn_readfirstlane(cg[6]);
            const int c7 = __builtin_amdgcn_readfirstlane(cg[7]);
            const int c8 = __builtin_amdgcn_readfirstlane(cg[8]);
            const int c9 = __builtin_amdgcn_readfirstlane(cg[9]);

            const float* w1n = w1 + (size_t)g * 40 + nn * 10;  // this lane's B column
            v8f acc = {};

            // -- k-step 0: lanes<16 use k={0,1}, lanes>=16 use k={2,3} --
            {
                v2f A, Bv;
                A.x = frow[half ? c2 : c0];
                A.y = frow[half ? c3 : c1];
                v2f Bw = *(const v2f*)(w1n + (half ? 2 : 0));   // unconditional b64
                Bv.x = nv ? Bw.x : 0.0f;
                Bv.y = nv ? Bw.y : 0.0f;
                acc = __builtin_amdgcn_wmma_f32_16x16x4_f32(
                          false, A, false, Bv, (short)0, acc, false, false);
            }
            // -- k-step 1: k={4,5} / {6,7} --
            {
                v2f A, Bv;
                A.x = frow[half ? c6 : c4];
                A.y = frow[half ? c7 : c5];
                v2f Bw = *(const v2f*)(w1n + (half ? 6 : 4));
                Bv.x = nv ? Bw.x : 0.0f;
                Bv.y = nv ? Bw.y : 0.0f;
                acc = __builtin_amdgcn_wmma_f32_16x16x4_f32(
                          false, A, false, Bv, (short)0, acc, false, false);
            }
            // -- k-step 2: k={8,9} for lanes<16; zero pad (k=10,11) for lanes>=16 --
            {
                v2f A, Bv;
                const float a0 = frow[c8];           // unconditional ds_load
                const float a1 = frow[c9];
                A.x = half ? 0.0f : a0;
                A.y = half ? 0.0f : a1;
                v2f Bw = *(const v2f*)(w1n + 8);     // unconditional b64 (idx<=39)
                const bool bok = nv && !half;
                Bv.x = bok ? Bw.x : 0.0f;
                Bv.y = bok ? Bw.y : 0.0f;
                acc = __builtin_amdgcn_wmma_f32_16x16x4_f32(
                          false, A, false, Bv, (short)0, acc, false, false);
            }

            // Epilogue: bias + leaky + w2 scaling; each n<4 lane owns a
            // contiguous m-range (m = 8*half .. 8*half+7) -> 2x ds_store_b128.
            if (nv) {
                const float b1v = b1[(size_t)g * 4 + nn];
                const float w2v = w2[(size_t)g * 4 + nn];
                v4f p0, p1;
                #pragma unroll
                for (int r = 0; r < 4; ++r) {
                    float t = acc[r] + b1v;
                    p0[r] = fmaxf(t, 0.01f * t) * w2v;
                }
                #pragma unroll
                for (int r = 0; r < 4; ++r) {
                    float t = acc[r + 4] + b1v;
                    p1[r] = fmaxf(t, 0.01f * t) * w2v;
                }
                float* dst = partial + gl * PSTRIDE + n * 16 + (half ? 8 : 0);
                *(v4f*)(dst)     = p0;               // m = 8h+0..3
                *(v4f*)(dst + 4) = p1;               // m = 8h+4..7
            }
        }
    }
    __syncthreads();

    // ---- n-reduction + b2 + coalesced writeback (64 consecutive genes/row) ----
    const int brow0 = btile * BT;
    for (int idx = tid; idx < BT * GPB; idx += NTHREADS) {
        const int m = idx >> 6;                     // batch row within tile
        const int j = idx & (GPB - 1);              // gene slot
        const int g = gblock + j;
        if (g < NUM_GENES) {
            const float* p = partial + j * PSTRIDE + m;
            const float s = (p[0] + p[16]) + (p[32] + p[48]);
            out[(size_t)(brow0 + m) * NUM_GENES + g] = s + b2[g];
        }
    }
}

extern "C" void kernel_launch(void* const* d_in, const int* in_sizes, int n_in,
                              void* d_out, int out_size, void* d_ws, size_t ws_size,
                              hipStream_t stream) {
    (void)in_sizes; (void)n_in; (void)out_size; (void)d_ws; (void)ws_size;
    const float* feat  = (const float*)d_in[0];
    const float* w1    = (const float*)d_in[1];
    const float* b1    = (const float*)d_in[2];
    const float* w2    = (const float*)d_in[3];
    const float* b2    = (const float*)d_in[4];
    const int*   cols1 = (const int*)  d_in[6];   // [5]=rows1 unused (structure implicit)
    float* out = (float*)d_out;

    dim3 grid((NUM_GENES + GPB - 1) / GPB, BATCH / BT, 1);
    gene_sparse2_kernel<<<grid, NTHREADS, 0, stream>>>(feat, w1, b1, w2, b2, cols1, out);
}